// RhythmMemoryUpdater_26293789786511
// MI455X (gfx1250) — compile-verified
//
#include <hip/hip_runtime.h>

typedef __attribute__((ext_vector_type(2))) float v2f;
typedef __attribute__((ext_vector_type(8))) float v8f;

#define MSG_DIM 128
#define MEM_DIM 128
#define CDIM    256
#define LDS_STRIDE 258   // 256 + 2 pad: row stride mod 64 banks == 2 -> conflict-free b64 loads

// ---------------------------------------------------------------------------
// Kernel 1: bulk copy node_memories -> out (512 MB; pure bandwidth, float4)
// ---------------------------------------------------------------------------
__global__ void rmu_copy_kernel(const float4* __restrict__ src,
                                float4* __restrict__ dst, long n4) {
    long i = (long)blockIdx.x * blockDim.x + threadIdx.x;
    long stride = (long)gridDim.x * blockDim.x;
    for (; i < n4; i += stride) dst[i] = src[i];
}

// ---------------------------------------------------------------------------
// Kernel 2: fused  (x * s) @ lin_w^T + b  ->  LayerNorm  ->  scatter
// 256 threads = 8 waves; wave w handles rows [blk*128 + w*16, +16), all 128 cols.
// GEMM via V_WMMA_F32_16X16X4_F32, weights (folded with conv scale) staged in LDS.
// ---------------------------------------------------------------------------
__global__ __launch_bounds__(256) void rmu_gemm_ln_scatter(
    const int*   __restrict__ node_ids,
    const float* __restrict__ messages,       // [B, 128]
    const float* __restrict__ node_memories,  // [N, 128]
    const float* __restrict__ conv_w,         // [256, 7]
    const float* __restrict__ lin_w,          // [128, 256]
    const float* __restrict__ lin_b,          // [128]
    const float* __restrict__ ln_gamma,       // [128]
    const float* __restrict__ ln_beta,        // [128]
    float*       __restrict__ out)            // [N, 128]
{
    __shared__ float ldsW[MEM_DIM * LDS_STRIDE];  // ~129 KB

    const int tid = threadIdx.x;

    // Stage folded weights: Wf[d][c] = lin_w[d][c] * conv_w[c][3]
    for (int i = tid; i < MEM_DIM * CDIM; i += 256) {
        const int d = i >> 8;
        const int c = i & 255;
        ldsW[d * LDS_STRIDE + c] = lin_w[i] * conv_w[c * 7 + 3];
    }
    __syncthreads();

    const int lane = tid & 31;
    const int wave = tid >> 5;
    const int half = lane >> 4;    // lane group: 0 -> lanes 0-15, 1 -> lanes 16-31
    const int l16  = lane & 15;

    const long rowBase = (long)blockIdx.x * 128 + (long)wave * 16;
    const long rowA    = rowBase + l16;   // this lane's A-matrix row (M = l%16)

    v8f acc[8] = {};   // 8 N-tiles of 16 -> full 128 output columns

    // A fragment addressing: lane holds A[M][k0 + 2*half + {0,1}] per k-step.
    const float* aptr1 = messages + rowA * MSG_DIM + 2 * half;
    const int    nidA  = node_ids[rowA];
    const float* aptr2 = node_memories + (long)nidA * MEM_DIM + 2 * half;

    // Phase 1: K = 0..127 (messages half of the concat)
    for (int kk = 0; kk < MSG_DIM; kk += 4) {
        v2f a = *(const v2f*)(aptr1 + kk);
        const float* wbase = &ldsW[l16 * LDS_STRIDE + kk + 2 * half];
#pragma unroll
        for (int nt = 0; nt < 8; ++nt) {
            v2f b = *(const v2f*)(wbase + nt * 16 * LDS_STRIDE);
            acc[nt] = __builtin_amdgcn_wmma_f32_16x16x4_f32(
                false, a, false, b, (short)0, acc[nt], false, false);
        }
    }
    // Phase 2: K = 128..255 (gathered old memory half of the concat)
    for (int kk = 0; kk < MEM_DIM; kk += 4) {
        v2f a = *(const v2f*)(aptr2 + kk);
        const float* wbase = &ldsW[l16 * LDS_STRIDE + (MSG_DIM + kk) + 2 * half];
#pragma unroll
        for (int nt = 0; nt < 8; ++nt) {
            v2f b = *(const v2f*)(wbase + nt * 16 * LDS_STRIDE);
            acc[nt] = __builtin_amdgcn_wmma_f32_16x16x4_f32(
                false, a, false, b, (short)0, acc[nt], false, false);
        }
    }

    // Epilogue params for this lane's columns d = nt*16 + l16
    float bias[8], gm[8], bt[8];
#pragma unroll
    for (int nt = 0; nt < 8; ++nt) {
        const int d = nt * 16 + l16;
        bias[nt] = lin_b[d];
        gm[nt]   = ln_gamma[d];
        bt[nt]   = ln_beta[d];
    }

    // C/D layout: element r of acc, lane l  ->  row M = r + 8*half, col N = l16.
    // Each output row lives across 16 lanes (one half) x 8 fragments.
#pragma unroll
    for (int r = 0; r < 8; ++r) {
        float vals[8];
        float s = 0.f;
#pragma unroll
        for (int nt = 0; nt < 8; ++nt) {
            vals[nt] = acc[nt][r] + bias[nt];
            s += vals[nt];
        }
        // reduce across the 16 lanes of this half (xor masks 1,2,4,8 stay in-half)
        for (int off = 1; off < 16; off <<= 1) s += __shfl_xor(s, off, 32);
        const float mean = s * (1.0f / 128.0f);

        float q = 0.f;
#pragma unroll
        for (int nt = 0; nt < 8; ++nt) {
            const float dlt = vals[nt] - mean;
            q += dlt * dlt;
        }
        for (int off = 1; off < 16; off <<= 1) q += __shfl_xor(q, off, 32);
        const float rstd = rsqrtf(q * (1.0f / 128.0f) + 1e-5f);

        const long row  = rowBase + r + 8 * half;
        const int  nid  = node_ids[row];
        float* obase = out + (long)nid * MEM_DIM;
#pragma unroll
        for (int nt = 0; nt < 8; ++nt) {
            obase[nt * 16 + l16] = (vals[nt] - mean) * rstd * gm[nt] + bt[nt];
        }
    }
}

// ---------------------------------------------------------------------------
extern "C" void kernel_launch(void* const* d_in, const int* in_sizes, int n_in,
                              void* d_out, int out_size, void* d_ws, size_t ws_size,
                              hipStream_t stream) {
    const int*   node_ids      = (const int*)  d_in[0];
    const float* messages      = (const float*)d_in[1];
    const float* node_memories = (const float*)d_in[2];
    const float* conv_w        = (const float*)d_in[3];
    const float* lin_w         = (const float*)d_in[4];
    const float* lin_b         = (const float*)d_in[5];
    const float* ln_gamma      = (const float*)d_in[6];
    const float* ln_beta       = (const float*)d_in[7];
    float*       out           = (float*)d_out;

    // 1) Full copy of node_memories into out (rows touched by node_ids get
    //    overwritten by kernel 2; both on `stream`, so ordering is guaranteed).
    const long n  = (long)in_sizes[2];   // NUM_NODES * MEM_DIM (multiple of 4)
    const long n4 = n >> 2;
    rmu_copy_kernel<<<16384, 256, 0, stream>>>(
        (const float4*)node_memories, (float4*)out, n4);

    // 2) Fused GEMM + LayerNorm + scatter: 128 rows per 256-thread block.
    const int nrows  = in_sizes[0];          // B = 262144
    const int blocks = nrows / 128;          // 2048
    rmu_gemm_ln_scatter<<<blocks, 256, 0, stream>>>(
        node_ids, messages, node_memories, conv_w, lin_w, lin_b,
        ln_gamma, ln_beta, out);
}